// PermLayer_31576599561039
// MI455X (gfx1250) — compile-verified
//
#include <hip/hip_runtime.h>
#include <stdint.h>

#define EMB_N 512
#define ROWS_PER_BLOCK 8

typedef __attribute__((ext_vector_type(4))) float v4f;
// Match the builtin's parameter pointee type exactly (GCC vector of 4 ints).
typedef int v4i __attribute__((vector_size(16)));

#if defined(__gfx1250__) && __has_builtin(__builtin_amdgcn_global_load_async_to_lds_b128) && __has_builtin(__builtin_amdgcn_s_wait_asynccnt)
#define USE_ASYNC_LDS 1
#else
#define USE_ASYNC_LDS 0
#endif

#if USE_ASYNC_LDS
typedef __attribute__((address_space(1))) v4i g_v4i;  // global int4
typedef __attribute__((address_space(3))) v4i l_v4i;  // LDS int4
// Generic->AS casts via integer round trip. For LDS, the low 32 bits of a
// generic pointer are the LDS byte offset (aperture lives in the high bits).
__device__ __forceinline__ g_v4i* to_global_v4i(const void* p) {
  return (g_v4i*)(uintptr_t)p;
}
__device__ __forceinline__ l_v4i* to_lds_v4i(const void* p) {
  return (l_v4i*)(uintptr_t)(uint32_t)(uintptr_t)p;
}
#endif

// Pass 1: recover the inverse permutation from the one-hot permKey rows.
// permKey[k, j] == 1  =>  out column j comes from input column k.
__global__ void __launch_bounds__(256) build_inv_kernel(
    const float* __restrict__ permKey, int* __restrict__ inv) {
  const int idx = blockIdx.x * 256 + threadIdx.x;
  if (idx < EMB_N * EMB_N) {
    if (permKey[idx] != 0.0f) {
      inv[idx & (EMB_N - 1)] = idx >> 9;  // inv[col] = row
    }
  }
}

// Pass 2: out[b, j] = x[b, inv[j]].  One wave per row: async-stage the row
// into LDS with b128 DMA, gather through LDS, store coalesced float4s.
__global__ void __launch_bounds__(256) perm_gather_kernel(
    const float* __restrict__ x, const int* __restrict__ inv,
    float* __restrict__ out, int nrows) {
  __shared__ int s_inv[EMB_N];
  __shared__ __align__(16) float s_row[ROWS_PER_BLOCK][EMB_N];

  const int tid  = threadIdx.x;
  const int wave = tid >> 5;   // 0..7
  const int lane = tid & 31;

  for (int i = tid; i < EMB_N; i += 256) s_inv[i] = inv[i];
  __syncthreads();

  for (long long base = (long long)blockIdx.x * ROWS_PER_BLOCK; base < nrows;
       base += (long long)gridDim.x * ROWS_PER_BLOCK) {
    const long long row = base + wave;       // uniform per wave -> full EXEC
    if (row < nrows) {
      const float* src = x + row * EMB_N;
      float*       dst = out + row * EMB_N;

#if USE_ASYNC_LDS
      // 4 issues x 32 lanes x 16B = 2KB row staged into LDS, no VGPR transit.
      for (int q = 0; q < 4; ++q) {
        const int elt = (q * 32 + lane) * 4;
        __builtin_amdgcn_global_load_async_to_lds_b128(
            to_global_v4i(src + elt), to_lds_v4i(&s_row[wave][elt]), 0, 0);
      }
      __builtin_amdgcn_s_wait_asynccnt(0);
      asm volatile("" ::: "memory");
#else
      for (int q = 0; q < 4; ++q) {
        const int elt = (q * 32 + lane) * 4;
        *(v4f*)&s_row[wave][elt] = *(const v4f*)(src + elt);
      }
      asm volatile("" ::: "memory");  // same-wave LDS RAW handled via DScnt
#endif

      // Permuted gather from LDS, fully coalesced b128 stores to global.
      for (int c = 0; c < 4; ++c) {
        const int j = c * 128 + lane * 4;
        v4f v;
        v.x = s_row[wave][s_inv[j + 0]];
        v.y = s_row[wave][s_inv[j + 1]];
        v.z = s_row[wave][s_inv[j + 2]];
        v.w = s_row[wave][s_inv[j + 3]];
        *(v4f*)(dst + j) = v;
      }
    }
  }
}

extern "C" void kernel_launch(void* const* d_in, const int* in_sizes, int n_in,
                              void* d_out, int out_size, void* d_ws, size_t ws_size,
                              hipStream_t stream) {
  const float* x       = (const float*)d_in[0];
  const float* permKey = (const float*)d_in[1];
  float*       out     = (float*)d_out;
  int*         inv     = (int*)d_ws;          // 512 ints of scratch
  const int nrows      = in_sizes[0] / EMB_N; // 65536

  build_inv_kernel<<<(EMB_N * EMB_N) / 256, 256, 0, stream>>>(permKey, inv);

  const int grid = (nrows + ROWS_PER_BLOCK - 1) / ROWS_PER_BLOCK; // 8192 blocks
  perm_gather_kernel<<<grid, 256, 0, stream>>>(x, inv, out, nrows);
}